// GCN_52484500357541
// MI455X (gfx1250) — compile-verified
//
#include <hip/hip_runtime.h>

#define NNODES 50000
#define NEDGES 800000
#define OUTC   128
#define NEG_SLOPE 0.2f

typedef __attribute__((ext_vector_type(2))) float v2f;
typedef __attribute__((ext_vector_type(8))) float v8f;

__global__ void zero_f32(float* __restrict__ p, int n) {
  int i = blockIdx.x * blockDim.x + threadIdx.x;
  if (i < n) p[i] = 0.0f;
}

__global__ void count_deg(const int* __restrict__ cols, float* __restrict__ deg, int E) {
  int i = blockIdx.x * blockDim.x + threadIdx.x;
  if (i < E) atomicAdd(&deg[cols[i]], 1.0f);
}

// C[N,128] = A[N,K] @ B[K,128] via V_WMMA_F32_16X16X4_F32.
// Block = 256 threads = 8 waves; one block handles a 16-row strip, each wave
// one 16x16 output tile (8 col tiles cover all 128 cols). A strip staged in
// LDS with padded stride (K+4) to avoid 64-bank conflicts when K%64==0.
__global__ void __launch_bounds__(256) gemm_wmma_f32(
    const float* __restrict__ A, const float* __restrict__ B,
    float* __restrict__ C, int K) {
  __shared__ float As[16 * 304];            // max K=300, stride K+4
  const int Ks = K + 4;
  const int row0 = blockIdx.x * 16;
  for (int i = threadIdx.x; i < 16 * K; i += 256) {
    int m = i / K, k = i - m * K;
    As[m * Ks + k] = A[(size_t)(row0 + m) * K + k];
  }
  __syncthreads();

  const int wave = threadIdx.x >> 5;
  const int lane = threadIdx.x & 31;
  const int lh   = lane & 15;
  const int hi   = lane >> 4;               // 0: lanes 0-15, 1: lanes 16-31
  const int col0 = wave * 16;

  v8f acc = {};
  for (int k0 = 0; k0 < K; k0 += 4) {
    v2f a, b;
    // A 16x4 frag: lanes 0-15 -> K=k0,k0+1 ; lanes 16-31 -> K=k0+2,k0+3
    a.x = As[lh * Ks + k0 + 2 * hi + 0];
    a.y = As[lh * Ks + k0 + 2 * hi + 1];
    // B 4x16 frag: same K split across lane halves, N = col0+lh
    b.x = B[(size_t)(k0 + 2 * hi + 0) * OUTC + col0 + lh];
    b.y = B[(size_t)(k0 + 2 * hi + 1) * OUTC + col0 + lh];
    acc = __builtin_amdgcn_wmma_f32_16x16x4_f32(
        /*neg_a=*/false, a, /*neg_b=*/false, b,
        /*c_mod=*/(short)0, acc, /*reuse_a=*/false, /*reuse_b=*/false);
  }

  // C/D layout: VGPR i -> M = i + 8*hi, N = lh
  for (int i = 0; i < 8; ++i) {
    int m = i + 8 * hi;
    C[(size_t)(row0 + m) * OUTC + col0 + lh] = acc[i];
  }
}

// One wave per edge: 32 lanes x float4 = one full 128-float row.
// Gather source row (L2-resident), scatter-add into agg[c] with f32 atomics.
__global__ void __launch_bounds__(256) scatter_edges(
    const float* __restrict__ hw, const int* __restrict__ rows,
    const int* __restrict__ cols, float* __restrict__ agg, int E) {
  int e = blockIdx.x * 8 + (threadIdx.x >> 5);
  if (e >= E) return;
  int lane = threadIdx.x & 31;
  int r = rows[e], c = cols[e];
  const float4 v = ((const float4*)(hw + (size_t)r * OUTC))[lane];
  float* dst = agg + (size_t)c * OUTC + lane * 4;
  atomicAdd(dst + 0, v.x);
  atomicAdd(dst + 1, v.y);
  atomicAdd(dst + 2, v.z);
  atomicAdd(dst + 3, v.w);
}

// out = leaky_relu(agg * inv_deg + bias) (relu optional on last layer)
__global__ void finalize_layer(const float* __restrict__ agg,
                               const float* __restrict__ deg,
                               const float* __restrict__ bias,
                               float* __restrict__ out, int relu) {
  int idx = blockIdx.x * blockDim.x + threadIdx.x;
  if (idx >= NNODES * OUTC) return;
  int rowi = idx >> 7;
  int c = idx & (OUTC - 1);
  float d = deg[rowi];
  float inv = d > 0.0f ? 1.0f / d : 0.0f;
  float v = fmaf(agg[idx], inv, bias[c]);
  if (relu) v = v > 0.0f ? v : NEG_SLOPE * v;
  out[idx] = v;
}

// Row-wise L2 normalize: one wave per row, float4 per lane, wave32 shuffle reduce.
__global__ void __launch_bounds__(256) l2_normalize(const float* __restrict__ h,
                                                    float* __restrict__ out) {
  int row = blockIdx.x * 8 + (threadIdx.x >> 5);
  if (row >= NNODES) return;
  int lane = threadIdx.x & 31;
  float4 v = ((const float4*)(h + (size_t)row * OUTC))[lane];
  float s = v.x * v.x + v.y * v.y + v.z * v.z + v.w * v.w;
  for (int off = 16; off > 0; off >>= 1) s += __shfl_xor(s, off, 32);
  float scale = 1.0f / fmaxf(sqrtf(s), 1e-12f);
  float4 o;
  o.x = v.x * scale; o.y = v.y * scale; o.z = v.z * scale; o.w = v.w * scale;
  ((float4*)(out + (size_t)row * OUTC))[lane] = o;
}

extern "C" void kernel_launch(void* const* d_in, const int* in_sizes, int n_in,
                              void* d_out, int out_size, void* d_ws, size_t ws_size,
                              hipStream_t stream) {
  const float* x  = (const float*)d_in[0];
  const int*   er = (const int*)d_in[1];
  const int*   ec = (const int*)d_in[2];
  const float* w1 = (const float*)d_in[3];
  const float* b1 = (const float*)d_in[4];
  const float* w2 = (const float*)d_in[5];
  const float* b2 = (const float*)d_in[6];
  const float* w3 = (const float*)d_in[7];
  const float* b3 = (const float*)d_in[8];
  float* out = (float*)d_out;

  float* ws  = (float*)d_ws;
  float* deg = ws;                                   // NNODES floats
  float* hw  = ws + 50048;                           // NNODES*128 (16B-aligned)
  float* agg = hw + (size_t)NNODES * OUTC;           // NNODES*128
  float* h   = agg + (size_t)NNODES * OUTC;          // NNODES*128

  const int NM = NNODES * OUTC;
  const int ZB = (NM + 255) / 256;
  const int EB = (NEDGES + 7) / 8;                   // 8 waves (edges) per block
  const int GB = NNODES / 16;                        // 50000/16 = 3125 exactly
  const int RB = (NNODES + 7) / 8;

  // Degree (re-zero every call; ws is poisoned by harness)
  zero_f32<<<(NNODES + 255) / 256, 256, 0, stream>>>(deg, NNODES);
  count_deg<<<(NEDGES + 255) / 256, 256, 0, stream>>>(ec, deg, NEDGES);

  // Layer 1: K=300, relu
  gemm_wmma_f32<<<GB, 256, 0, stream>>>(x, w1, hw, 300);
  zero_f32<<<ZB, 256, 0, stream>>>(agg, NM);
  scatter_edges<<<EB, 256, 0, stream>>>(hw, er, ec, agg, NEDGES);
  finalize_layer<<<ZB, 256, 0, stream>>>(agg, deg, b1, h, 1);

  // Layer 2: K=128, relu
  gemm_wmma_f32<<<GB, 256, 0, stream>>>(h, w2, hw, 128);
  zero_f32<<<ZB, 256, 0, stream>>>(agg, NM);
  scatter_edges<<<EB, 256, 0, stream>>>(hw, er, ec, agg, NEDGES);
  finalize_layer<<<ZB, 256, 0, stream>>>(agg, deg, b2, h, 1);

  // Layer 3: K=128, no relu
  gemm_wmma_f32<<<GB, 256, 0, stream>>>(h, w3, hw, 128);
  zero_f32<<<ZB, 256, 0, stream>>>(agg, NM);
  scatter_edges<<<EB, 256, 0, stream>>>(hw, er, ec, agg, NEDGES);
  finalize_layer<<<ZB, 256, 0, stream>>>(agg, deg, b3, h, 0);

  // Row-wise L2 normalize into output
  l2_normalize<<<RB, 256, 0, stream>>>(h, out);
}